// PyTorchModel_50543175139385
// MI455X (gfx1250) — compile-verified
//
#include <hip/hip_runtime.h>

typedef float v2f __attribute__((ext_vector_type(2)));
typedef float v8f __attribute__((ext_vector_type(8)));

#define HIDDEN   50
#define INPUT    4
#define TSTEPS   512
#define KP       64          // padded K: 50 h | 2 pad | 4 x | 1 bias-one | 7 pad
#define GP       256         // padded gates: 4 blocks of 64 (i,f,g,o)
#define STR      68          // LDS column stride in floats (bank-conflict padding)
#define XCOL     52          // x_t lives in A columns 52..55 (16B aligned)
#define BCOL     56          // constant-1 column (bias row of fused W)
#define ROWS     16          // batch rows per wave (WMMA M)
#define WAVES    2           // waves per workgroup sharing one weight copy
#define WFLOATS  (GP * STR)  // 17408 floats = 68 KB fused-weight tile
#define HFLOATS  (ROWS * STR)

// sigmoid: v_exp + v_add + v_rcp (approx rcp, no full-precision div sequence)
__device__ __forceinline__ float sigm(float v) {
  return __builtin_amdgcn_rcpf(1.0f + __expf(-v));
}

// tanh: single V_TANH_F32 on CDNA5 if the builtin exists; exp/rcp fallback
__device__ __forceinline__ float tanh_f(float v) {
#if __has_builtin(__builtin_amdgcn_tanh_f32)
  return __builtin_amdgcn_tanh_f32(v);
#elif __has_builtin(__builtin_amdgcn_tanhf)
  return __builtin_amdgcn_tanhf(v);
#else
  v = fminf(fmaxf(v, -15.0f), 15.0f);
  const float e = __expf(2.0f * v);
  return (e - 1.0f) * __builtin_amdgcn_rcpf(e + 1.0f);
#endif
}

__global__ __launch_bounds__(32 * WAVES)
void lstm_wmma_kernel(const float* __restrict__ x,
                      const float* __restrict__ W_ih,
                      const float* __restrict__ W_hh,
                      const float* __restrict__ b_ih,
                      const float* __restrict__ b_hh,
                      const float* __restrict__ W_fc,
                      const float* __restrict__ b_fc,
                      float* __restrict__ out)
{
  extern __shared__ float smem[];
  float* Wl = smem;                                  // [GP][STR] fused weights
  const int lane = threadIdx.x & 31;
  const int wave = threadIdx.x >> 5;
  float* hl = smem + WFLOATS + wave * HFLOATS;       // [ROWS][STR] A-matrix tile

  // ---- build fused/padded weight matrix in LDS: Wl[col][k] ----
  for (int idx = threadIdx.x; idx < GP * KP; idx += 32 * WAVES) {
    const int pc = idx >> 6;
    const int k  = idx & 63;
    const int q  = pc >> 6;
    const int i  = pc & 63;
    float v = 0.0f;
    if (i < HIDDEN) {
      const int g = q * HIDDEN + i;
      if (k < HIDDEN)                          v = W_hh[g * HIDDEN + k];
      else if (k >= XCOL && k < XCOL + INPUT)  v = W_ih[g * INPUT + (k - XCOL)];
      else if (k == BCOL)                      v = b_ih[g] + b_hh[g];
    }
    Wl[pc * STR + k] = v;
  }

  // ---- init per-wave A tile: h0 = 0, constant-1 column, x at t=0 ----
  for (int idx = lane; idx < HFLOATS; idx += 32) hl[idx] = 0.0f;
  const int row0 = blockIdx.x * (WAVES * ROWS) + wave * ROWS;
  if (lane < ROWS) {
    hl[lane * STR + BCOL] = 1.0f;
    const float4 x0 = *(const float4*)(x + (size_t)(row0 + lane) * TSTEPS * INPUT);
    *(float4*)(&hl[lane * STR + XCOL]) = x0;
  }
  __syncthreads();

  const int n0   = lane & 15;          // A row / B column-within-tile
  const int koff = (lane >> 4) * 2;    // K sub-offset for upper half-wave
  const int mrow = (lane >> 4) * 8;    // D-layout M base for this lane

  v8f cc[4];                           // cell state, resident 512 steps
  v8f hn[4];                           // h_new in D layout
  #pragma unroll
  for (int j = 0; j < 4; ++j)
    #pragma unroll
    for (int r = 0; r < 8; ++r) { cc[j][r] = 0.0f; hn[j][r] = 0.0f; }

  #pragma unroll 1
  for (int t = 0; t < TSTEPS; ++t) {
    // A fragments: lane<16 holds (M=n0, K=4kk..4kk+1), lane>=16 (K=4kk+2..+3)
    v2f A[16];
    #pragma unroll
    for (int kk = 0; kk < 16; ++kk)
      A[kk] = *(const v2f*)(&hl[n0 * STR + kk * 4 + koff]);

    // branch-free prefetch of next x (clamped index; upper half duplicates)
    const int tn = (t + 1 < TSTEPS) ? t + 1 : t;
    const float4 xn = *(const float4*)(x + ((size_t)(row0 + n0) * TSTEPS + tn) * INPUT);

    // gates[16x256] = A @ W̄: 8 interleaved WMMA accumulation chains
    #pragma unroll
    for (int jp = 0; jp < 2; ++jp) {
      v8f acc[8];
      #pragma unroll
      for (int q = 0; q < 8; ++q)
        #pragma unroll
        for (int r = 0; r < 8; ++r) acc[q][r] = 0.0f;

      int base[8];
      #pragma unroll
      for (int jj = 0; jj < 2; ++jj) {
        const int j = jp * 2 + jj;
        #pragma unroll
        for (int q = 0; q < 4; ++q)
          base[jj * 4 + q] = (q * 64 + j * 16 + n0) * STR + koff;
      }

      #pragma unroll
      for (int kk = 0; kk < 16; ++kk) {
        const v2f a = A[kk];
        #pragma unroll
        for (int q = 0; q < 8; ++q) {
          const v2f b = *(const v2f*)(&Wl[base[q] + kk * 4]);
          acc[q] = __builtin_amdgcn_wmma_f32_16x16x4_f32(
              false, a, false, b, (short)0, acc[q], false, false);
        }
      }

      // LSTM cell, elementwise in D layout (TRANS ops co-execute with VALU)
      #pragma unroll
      for (int jj = 0; jj < 2; ++jj) {
        const int j = jp * 2 + jj;
        #pragma unroll
        for (int r = 0; r < 8; ++r) {
          const float iv = sigm(acc[jj * 4 + 0][r]);
          const float fv = sigm(acc[jj * 4 + 1][r]);
          const float gv = tanh_f(acc[jj * 4 + 2][r]);
          const float ov = sigm(acc[jj * 4 + 3][r]);
          const float cn = fv * cc[j][r] + iv * gv;
          cc[j][r] = cn;
          hn[j][r] = ov * tanh_f(cn);
        }
      }
    }

    // write h_new back in A layout (cols 0..49; x/bias cols preserved)
    #pragma unroll
    for (int j = 0; j < 4; ++j) {
      const int n = j * 16 + n0;
      if (n < HIDDEN) {
        #pragma unroll
        for (int r = 0; r < 8; ++r)
          hl[(mrow + r) * STR + n] = hn[j][r];
      }
    }
    *(float4*)(&hl[n0 * STR + XCOL]) = xn;  // both half-waves write same data
  }

  // ---- out[b] = h_T . W_fc + b_fc ----
  float wf[4];
  #pragma unroll
  for (int j = 0; j < 4; ++j) {
    const int n = j * 16 + n0;
    wf[j] = (n < HIDDEN) ? W_fc[n] : 0.0f;
  }
  __syncthreads();                      // reuse hl as reduction scratch
  #pragma unroll
  for (int r = 0; r < 8; ++r) {
    const float p = hn[0][r] * wf[0] + hn[1][r] * wf[1]
                  + hn[2][r] * wf[2] + hn[3][r] * wf[3];
    hl[(mrow + r) * 17 + n0] = p;       // part[m][src_lane], stride-17 pad
  }
  __syncthreads();
  if (lane < ROWS) {
    float s = b_fc[0];
    #pragma unroll
    for (int l = 0; l < 16; ++l) s += hl[lane * 17 + l];
    out[row0 + lane] = s;
  }
}

extern "C" void kernel_launch(void* const* d_in, const int* in_sizes, int n_in,
                              void* d_out, int out_size, void* d_ws, size_t ws_size,
                              hipStream_t stream) {
  (void)in_sizes; (void)n_in; (void)out_size; (void)d_ws; (void)ws_size;
  const float* x    = (const float*)d_in[0];
  const float* W_ih = (const float*)d_in[1];
  const float* W_hh = (const float*)d_in[2];
  const float* b_ih = (const float*)d_in[3];
  const float* b_hh = (const float*)d_in[4];
  const float* W_fc = (const float*)d_in[5];
  const float* b_fc = (const float*)d_in[6];
  float* out = (float*)d_out;

  const int B = 4096;
  const int blocks = B / (WAVES * ROWS);                            // 128
  const size_t shmem = (WFLOATS + WAVES * HFLOATS) * sizeof(float); // 78336 B
  static bool attr_set = false;
  if (!attr_set) {
    (void)hipFuncSetAttribute((const void*)lstm_wmma_kernel,
                              hipFuncAttributeMaxDynamicSharedMemorySize,
                              (int)shmem);
    attr_set = true;
  }
  lstm_wmma_kernel<<<blocks, WAVES * 32, shmem, stream>>>(
      x, W_ih, W_hh, b_ih, b_hh, W_fc, b_fc, out);
}